// ModelNew_3556232922318
// MI455X (gfx1250) — compile-verified
//
#include <hip/hip_runtime.h>
#include <math.h>

// MI455X / gfx1250, wave32. The problem is HBM-store-bound (272MB output @
// 23.3 TB/s ~= 12us; conv math is only ~34 GFLOP), so: full f32 precision via
// V_WMMA_F32_16X16X4_F32, async global->LDS (ASYNCcnt) double-buffered weight
// staging, and non-temporal 128-bit output stores.

typedef float v2f __attribute__((ext_vector_type(2)));
typedef float v8f __attribute__((ext_vector_type(8)));
// 4-byte-aligned float4 for dword-aligned b128 stores (rows of 129 floats).
typedef float f4u __attribute__((ext_vector_type(4), aligned(4)));

#define NBATCH 32
#define CIN    64
#define COUT   128
#define HIN    64
#define HOUT   129
#define HH     (HOUT * HOUT)
#define PT     8                   // p rows per workgroup tile
#define QT     16                  // q cols per workgroup tile
#define XROWS  (PT + 3)            // 11 input rows (halo 3)
#define XCOLS  (QT + 3)            // 19 input cols
#define XCOLP  (XCOLS + 1)         // 20, padded for banks
#define XSTRIDE (XROWS * XCOLP)    // 220 floats per input channel slab
#define WSLICE  (16 * 2 * 128 * 2) // 8192 floats per tap: [c4][hi][m] x float2
#define STG_STRIDE 17              // staging row stride (128 ch x 17)

// ---------------------------------------------------------------------------
// Kernel 1: const[c] = sigmoid(2 * softmax(bias)[c])  (value at all odd sites)
// ---------------------------------------------------------------------------
__global__ void prep_const(const float* __restrict__ bias, float* __restrict__ cconst) {
    __shared__ float red[128];
    int t = threadIdx.x;            // 128 threads
    float b = bias[t];
    red[t] = b; __syncthreads();
    for (int s = 64; s > 0; s >>= 1) {
        if (t < s) red[t] = fmaxf(red[t], red[t + s]);
        __syncthreads();
    }
    float mx = red[0]; __syncthreads();
    float e = __expf(b - mx);
    red[t] = e; __syncthreads();
    for (int s = 64; s > 0; s >>= 1) {
        if (t < s) red[t] += red[t + s];
        __syncthreads();
    }
    float sm = e / red[0];
    cconst[t] = 1.0f / (1.0f + __expf(-2.0f * sm));
}

// ---------------------------------------------------------------------------
// Kernel 2: repack weight (128,64,4,4) into the WMMA f32 16x16x4 A layout:
//   Wt[ ((tap*16 + c4)*2 + hi)*128 + m ] = float2{ W[m][4c4+2hi], W[m][4c4+2hi+1] }
// so a lane's A operand is one aligned b64 LDS read.
// ---------------------------------------------------------------------------
__global__ void wreorder(const float* __restrict__ W, float* __restrict__ Wt) {
    int idx = blockIdx.x * blockDim.x + threadIdx.x;   // 131072 total floats
    if (idx >= 16 * WSLICE) return;
    int j   = idx & 1;
    int m   = (idx >> 1) & 127;
    int hi  = (idx >> 8) & 1;
    int c4  = (idx >> 9) & 15;
    int tap = (idx >> 13) & 15;
    int ic  = 4 * c4 + 2 * hi + j;
    Wt[idx] = W[(m * CIN + ic) * 16 + tap];
}

// ---------------------------------------------------------------------------
// Kernel 3: fill all odd output rows with the per-channel constant (NT b128).
// ---------------------------------------------------------------------------
__global__ void fill_odd(const float* __restrict__ cconst, float* __restrict__ out) {
    int nc = blockIdx.x;                 // 0..4095 = n*128 + c
    float v = cconst[nc & 127];
    f4u v4; v4.x = v; v4.y = v; v4.z = v; v4.w = v;
    float* base = out + (size_t)nc * HH;
    // 64 odd rows x 32 float4 + 1 tail each
    for (int idx = threadIdx.x; idx < 64 * 32; idx += blockDim.x) {
        int r = idx >> 5, i = idx & 31;
        __builtin_nontemporal_store(v4, (f4u*)(base + (size_t)(2 * r + 1) * HOUT + i * 4));
    }
    for (int r = threadIdx.x; r < 64; r += blockDim.x)
        __builtin_nontemporal_store(v, base + (size_t)(2 * r + 1) * HOUT + 128);
}

// ---------------------------------------------------------------------------
// Async global->LDS copy of one 32KB weight tap slice (8 b128 per thread).
// Tracked by ASYNCcnt; in-order completion => s_wait_asynccnt N gates buffers.
// ---------------------------------------------------------------------------
__device__ __forceinline__ void wprefetch(const float* __restrict__ wsrc,
                                          float* ldsbuf, int tid) {
    uint32_t lds_base = (uint32_t)(uintptr_t)ldsbuf;   // low 32 bits = LDS offset
    #pragma unroll
    for (int i = 0; i < 8; ++i) {
        int f = tid + 256 * i;                         // float4 index, 2048 total
        const float* g = wsrc + f * 4;
        uint32_t loff = lds_base + f * 16;
        asm volatile("global_load_async_to_lds_b128 %0, %1, off"
                     :: "v"(loff), "v"(g) : "memory");
    }
}

// ---------------------------------------------------------------------------
// Kernel 4: implicit-GEMM transposed conv on the even/even lattice + fused
// bias + channel softmax + sigmoid, writing complete even output rows.
// Block: 256 threads = 8 wave32. Wave w handles p = p0+w, all 128 channels.
// ---------------------------------------------------------------------------
__global__ void __launch_bounds__(256)
convt_gemm(const float* __restrict__ x, const float* __restrict__ Wt,
           const float* __restrict__ bias, const float* __restrict__ cconst,
           float* __restrict__ out) {
    extern __shared__ float smem[];
    float* lds_x  = smem;                       // 14080 f
    float* lds_w0 = lds_x + CIN * XSTRIDE;      // 8192 f (tap double-buffer A)
    float* lds_w1 = lds_w0 + WSLICE;            // 8192 f (tap double-buffer B)
    float* lds_b  = lds_w1 + WSLICE;            // 128 f
    float* lds_c  = lds_b + COUT;               // 128 f  -> total 30720 f
    float* stage  = smem;                       // 17408 f, aliases x+w0 (dead)

    const int tid    = threadIdx.x;
    const int lane   = tid & 31;
    const int wid    = tid >> 5;        // wave id 0..7
    const int lane15 = lane & 15;
    const int hi     = (lane >> 4) & 1; // half-wave K-split per ISA layout

    const int q0 = blockIdx.x * QT;     // 0,16,32,48,64
    const int p0 = blockIdx.y * PT;     // 0,8,...,64
    const int n  = blockIdx.z;

    if (tid < COUT) { lds_b[tid] = bias[tid]; lds_c[tid] = cconst[tid]; }

    // Kick off async weight staging for taps 0 and 1 (depth-2 pipeline).
    wprefetch(Wt, lds_w0, tid);
    wprefetch(Wt + WSLICE, lds_w1, tid);

    // Stage zero-padded x tile: [ic][ih_local 0..10][iw_local 0..19]
    const float* xn = x + (size_t)n * CIN * HIN * HIN;
    for (int idx = tid; idx < CIN * XSTRIDE; idx += 256) {
        int cc = idx % XCOLP;
        int t2 = idx / XCOLP;
        int r  = t2 % XROWS;
        int ic = t2 / XROWS;
        int ih = p0 - 3 + r;
        int iw = q0 - 3 + cc;
        float v = 0.0f;
        if (cc < XCOLS && ih >= 0 && ih < HIN && iw >= 0 && iw < HIN)
            v = xn[(size_t)ic * (HIN * HIN) + ih * HIN + iw];
        lds_x[idx] = v;
    }

    v8f acc[8];
    #pragma unroll
    for (int mt = 0; mt < 8; ++mt)
        #pragma unroll
        for (int k = 0; k < 8; ++k) acc[mt][k] = 0.0f;

    for (int tap = 0; tap < 16; ++tap) {
        // Buffer for tap is ready when at most the next tap's 8 ops remain.
        if (tap < 15) asm volatile("s_wait_asynccnt 0x8" ::: "memory");
        else          asm volatile("s_wait_asynccnt 0x0" ::: "memory");
        __syncthreads();

        const float* lds_w = (tap & 1) ? lds_w1 : lds_w0;
        const int kh = tap >> 2, kw = tap & 3;
        const int rl = wid + 3 - kh;               // 0..10
        const int cl = lane15 + 3 - kw;            // 0..18
        const float* xb = &lds_x[rl * XCOLP + cl];

        #pragma unroll
        for (int c4 = 0; c4 < 16; ++c4) {
            // B (K=4 x N=16 f32): lanes 0-15 hold K=0,1; lanes 16-31 hold K=2,3
            const int ic = 4 * c4 + 2 * hi;
            v2f b;
            b.x = xb[(ic + 0) * XSTRIDE];
            b.y = xb[(ic + 1) * XSTRIDE];
            const v2f* wp = (const v2f*)&lds_w[(c4 * 2 + hi) * 256];
            v2f a[8];
            #pragma unroll
            for (int mt = 0; mt < 8; ++mt) a[mt] = wp[mt * 16 + lane15];
            #pragma unroll
            for (int mt = 0; mt < 8; ++mt) {
#if __has_builtin(__builtin_amdgcn_wmma_f32_16x16x4_f32)
                acc[mt] = __builtin_amdgcn_wmma_f32_16x16x4_f32(
                    false, a[mt], false, b, (short)0, acc[mt], false, false);
#else
                acc[mt][0] += a[mt].x * b.x + a[mt].y * b.y;
#endif
            }
        }
        __syncthreads();   // all waves done with this buffer
        if (tap + 2 < 16)  // refill the just-freed buffer with tap+2
            wprefetch(Wt + (size_t)(tap + 2) * WSLICE, (tap & 1) ? lds_w1 : lds_w0, tid);
    }

    // Epilogue: bias, channel softmax (128 ch split across lane pair L/L+16),
    // sigmoid(2*softmax). Lane holds channels {16*mt + 8*hi + k}.
    #pragma unroll
    for (int mt = 0; mt < 8; ++mt)
        #pragma unroll
        for (int k = 0; k < 8; ++k)
            acc[mt][k] += lds_b[16 * mt + 8 * hi + k];

    float mx = -3.4e38f;
    #pragma unroll
    for (int mt = 0; mt < 8; ++mt)
        #pragma unroll
        for (int k = 0; k < 8; ++k) mx = fmaxf(mx, acc[mt][k]);
    mx = fmaxf(mx, __shfl_xor(mx, 16, 32));

    float sum = 0.0f;
    #pragma unroll
    for (int mt = 0; mt < 8; ++mt)
        #pragma unroll
        for (int k = 0; k < 8; ++k) {
            float e = __expf(acc[mt][k] - mx);
            acc[mt][k] = e;
            sum += e;
        }
    sum += __shfl_xor(sum, 16, 32);
    const float inv = 1.0f / sum;

    // stage[] aliases the (now dead, barriered-off) x/w0 regions.
    float* st = stage + wid * (COUT * STG_STRIDE);
    #pragma unroll
    for (int mt = 0; mt < 8; ++mt)
        #pragma unroll
        for (int k = 0; k < 8; ++k) {
            int c   = 16 * mt + 8 * hi + k;
            float s = acc[mt][k] * inv;
            st[c * STG_STRIDE + lane15] = 1.0f / (1.0f + __expf(-2.0f * s));
        }
    __syncthreads();

    // Store full even row h=2p: computed evens interleaved with the constant
    // at odd columns, as NT b128 (dword-aligned ok on gfx1250).
    const int p = p0 + wid;
    if (p <= 64) {
        const int h = 2 * p;
        if (q0 < 64) {                      // cols [2q0 .. 2q0+31], all < 128
            const int cj   = lane >> 3;     // channel sub-index 0..3
            const int col4 = (lane & 7) * 4;
            const int ql   = col4 >> 1;     // stage q index of first element
            size_t rowbase = ((size_t)n * COUT) * HH + (size_t)h * HOUT + 2 * q0 + col4;
            #pragma unroll 4
            for (int cb = 0; cb < COUT; cb += 4) {
                int c = cb + cj;
                float cv = lds_c[c];
                const float* sr = st + c * STG_STRIDE + ql;
                f4u v; v.x = sr[0]; v.y = cv; v.z = sr[1]; v.w = cv;
                __builtin_nontemporal_store(v, (f4u*)(out + rowbase + (size_t)c * HH));
            }
        } else {                            // q0 == 64: only column 128 valid
            for (int c = lane; c < COUT; c += 32)
                __builtin_nontemporal_store(
                    st[c * STG_STRIDE],
                    out + ((size_t)n * COUT + c) * HH + (size_t)h * HOUT + 128);
        }
    }
}

// ---------------------------------------------------------------------------
extern "C" void kernel_launch(void* const* d_in, const int* in_sizes, int n_in,
                              void* d_out, int out_size, void* d_ws, size_t ws_size,
                              hipStream_t stream) {
    const float* x    = (const float*)d_in[0];   // (32,64,64,64)
    const float* W    = (const float*)d_in[1];   // (128,64,4,4)
    const float* bias = (const float*)d_in[2];   // (128,)
    float* out = (float*)d_out;

    float* cconst = (float*)d_ws;                // 128 floats
    float* Wt     = cconst + 128;                // 131072 floats (512KB)

    size_t smem = (size_t)(CIN * XSTRIDE + 2 * WSLICE + 2 * COUT) * sizeof(float);
    // 122880 B -> 2 workgroups per WGP (320KB LDS)
    (void)hipFuncSetAttribute((const void*)convt_gemm,
                              hipFuncAttributeMaxDynamicSharedMemorySize, (int)smem);

    prep_const<<<1, 128, 0, stream>>>(bias, cconst);
    wreorder<<<(16 * WSLICE + 255) / 256, 256, 0, stream>>>(W, Wt);
    fill_odd<<<NBATCH * COUT, 256, 0, stream>>>(cconst, out);

    dim3 grid(5 /*q tiles*/, 9 /*p tiles*/, NBATCH);
    convt_gemm<<<grid, 256, smem, stream>>>(x, Wt, bias, cconst, out);
}